// GCN_40467181863493
// MI455X (gfx1250) — compile-verified
//
#include <hip/hip_runtime.h>
#include <hip/hip_bf16.h>

typedef __attribute__((ext_vector_type(2)))  float v2f;
typedef __attribute__((ext_vector_type(8)))  float v8f;

// ---------------------------------------------------------------------------
// Dense GEMM: H[M,64] = X[M,64] @ W[64,64], fp32, via V_WMMA_F32_16X16X4_F32.
// One wave (32 lanes) computes a 16x64 output tile; 16 K-steps of 4, 4 N-tiles.
//
// A 16x4 f32 layout: lanes 0-15 hold (M=lane, K=kb+0,kb+1) in a VGPR pair,
//                    lanes 16-31 hold (K=kb+2,kb+3).
// B 4x16 f32 layout: VGPR0 = row K=kb+0 (lanes 0-15) / K=kb+2 (lanes 16-31),
//                    VGPR1 = K=kb+1 / K=kb+3, lane%16 = N.
// W is staged into LDS PRE-SWIZZLED into that exact per-lane float2 layout so
// each B operand is one aligned ds_load_b64 straight into an even VGPR pair.
// sB index: [kk][ntile][lane] -> kk*128 + ntile*32 + lane  (float2 units)
// ---------------------------------------------------------------------------
__global__ __launch_bounds__(256) void gcn_gemm64(
    const float* __restrict__ X, const float* __restrict__ W,
    float* __restrict__ H, int M) {
  __shared__ float2 sB[16 * 4 * 32];            // 16 KB
  const int tid = threadIdx.x;
  // pre-swizzle W[64,64] into WMMA B-operand layout
  for (int idx = tid; idx < 16 * 4 * 32; idx += 256) {
    const int kk   = idx >> 7;
    const int rem  = idx & 127;
    const int nt   = rem >> 5;
    const int ln   = rem & 31;
    const int kb   = kk * 4 + (ln >> 4) * 2;    // K row for this lane's VGPR0
    const int n    = nt * 16 + (ln & 15);
    sB[idx] = make_float2(W[kb * 64 + n], W[(kb + 1) * 64 + n]);
  }
  __syncthreads();

  const int wave = tid >> 5;
  const int lane = tid & 31;
  const int tile = blockIdx.x * 8 + wave;       // 16-row tile index
  if (tile * 16 >= M) return;                   // wave-uniform: EXEC stays all-1s
  const int m0   = tile * 16;
  const int half = lane >> 4;
  const int l15  = lane & 15;

  const float* xrow = X + (size_t)(m0 + l15) * 64 + half * 2;
  const float2* bl  = sB + lane;

  v8f acc0 = {}, acc1 = {}, acc2 = {}, acc3 = {};
#pragma unroll
  for (int kk = 0; kk < 16; ++kk) {
    const float2 av = *(const float2*)(xrow + kk * 4);
    v2f a; a.x = av.x; a.y = av.y;
    const float2 bv0 = bl[kk * 128 + 0 * 32];
    const float2 bv1 = bl[kk * 128 + 1 * 32];
    const float2 bv2 = bl[kk * 128 + 2 * 32];
    const float2 bv3 = bl[kk * 128 + 3 * 32];
    v2f b0, b1, b2, b3;
    b0.x = bv0.x; b0.y = bv0.y;
    b1.x = bv1.x; b1.y = bv1.y;
    b2.x = bv2.x; b2.y = bv2.y;
    b3.x = bv3.x; b3.y = bv3.y;
    acc0 = __builtin_amdgcn_wmma_f32_16x16x4_f32(false, a, false, b0, (short)0, acc0, false, false);
    acc1 = __builtin_amdgcn_wmma_f32_16x16x4_f32(false, a, false, b1, (short)0, acc1, false, false);
    acc2 = __builtin_amdgcn_wmma_f32_16x16x4_f32(false, a, false, b2, (short)0, acc2, false, false);
    acc3 = __builtin_amdgcn_wmma_f32_16x16x4_f32(false, a, false, b3, (short)0, acc3, false, false);
  }

  // C/D layout: VGPR r -> row m0 + r + half*8, col = lane&15 (+16/32/48)
  float* hrow = H + (size_t)(m0 + half * 8) * 64 + l15;
#pragma unroll
  for (int r = 0; r < 8; ++r) {
    hrow[(size_t)r * 64 + 0]  = acc0[r];
    hrow[(size_t)r * 64 + 16] = acc1[r];
    hrow[(size_t)r * 64 + 32] = acc2[r];
    hrow[(size_t)r * 64 + 48] = acc3[r];
  }
}

// ---------------------------------------------------------------------------
// Graph / elementwise kernels
// ---------------------------------------------------------------------------
__global__ void k_fill(float* __restrict__ p, float v, int n) {
  int i = blockIdx.x * 256 + threadIdx.x;
  if (i < n) p[i] = v;
}

__global__ void k_deg(const long long* __restrict__ dst, float* __restrict__ deg, int E) {
  int i = blockIdx.x * 256 + threadIdx.x;
  if (i < E) atomicAdd(&deg[(int)dst[i]], 1.0f);
}

__global__ void k_rsqrt(float* __restrict__ d, int n) {
  int i = blockIdx.x * 256 + threadIdx.x;
  if (i < n) { float v = d[i]; d[i] = (v > 0.0f) ? rsqrtf(v) : 0.0f; }
}

// out[node,f] = h[node,f] * dinv[node]^2   (self-loop message)
__global__ void k_self(const float* __restrict__ h, const float* __restrict__ dinv,
                       float* __restrict__ out, int total) {
  int i = blockIdx.x * 256 + threadIdx.x;
  if (i < total) { float dv = dinv[i >> 6]; out[i] = h[i] * dv * dv; }
}

// one wave per edge; lane handles 2 consecutive feats (coalesced 256B/edge)
__global__ void k_scatter(const long long* __restrict__ src, const long long* __restrict__ dst,
                          const float* __restrict__ dinv, const float* __restrict__ h,
                          float* __restrict__ out, int E) {
  int gid  = blockIdx.x * 256 + threadIdx.x;
  int e    = gid >> 5;
  int lane = gid & 31;
  if (e >= E) return;
  int s = (int)src[e], d = (int)dst[e];
  float nrm = dinv[s] * dinv[d];
  float2 v = *(const float2*)(h + (size_t)s * 64 + lane * 2);
  atomicAdd(out + (size_t)d * 64 + lane * 2,     v.x * nrm);
  atomicAdd(out + (size_t)d * 64 + lane * 2 + 1, v.y * nrm);
}

__global__ void k_finish(float* __restrict__ out, const float* __restrict__ b,
                         int total, int relu) {
  int i = blockIdx.x * 256 + threadIdx.x;
  if (i < total) {
    float v = out[i] + b[i & 63];
    out[i] = relu ? fmaxf(v, 0.0f) : v;
  }
}

// one wave per node: segment-sum into [G,64] + counts
__global__ void k_pool(const float* __restrict__ f, const long long* __restrict__ batch,
                       float* __restrict__ psum, float* __restrict__ pcnt, int N) {
  int gid  = blockIdx.x * 256 + threadIdx.x;
  int node = gid >> 5;
  int lane = gid & 31;
  if (node >= N) return;
  int g = (int)batch[node];
  float2 v = *(const float2*)(f + (size_t)node * 64 + lane * 2);
  atomicAdd(psum + (size_t)g * 64 + lane * 2,     v.x);
  atomicAdd(psum + (size_t)g * 64 + lane * 2 + 1, v.y);
  if (lane == 0) atomicAdd(&pcnt[g], 1.0f);
}

// one wave per graph: mean, 64-dot with Wl, + bl, sigmoid
__global__ void k_head(const float* __restrict__ psum, const float* __restrict__ pcnt,
                       const float* __restrict__ Wl, const float* __restrict__ bl,
                       float* __restrict__ out, int G) {
  int gid  = blockIdx.x * 256 + threadIdx.x;
  int g    = gid >> 5;
  int lane = gid & 31;
  if (g >= G) return;
  float c = fmaxf(pcnt[g], 1.0f);
  float2 s = *(const float2*)(psum + (size_t)g * 64 + lane * 2);
  float p = (s.x * Wl[lane * 2] + s.y * Wl[lane * 2 + 1]) / c;
  for (int off = 16; off > 0; off >>= 1) p += __shfl_down(p, off, 32);
  if (lane == 0) out[g] = 1.0f / (1.0f + __expf(-(p + bl[0])));
}

// ---------------------------------------------------------------------------
extern "C" void kernel_launch(void* const* d_in, const int* in_sizes, int n_in,
                              void* d_out, int out_size, void* d_ws, size_t ws_size,
                              hipStream_t stream) {
  const float*     x     = (const float*)d_in[0];
  const long long* ei    = (const long long*)d_in[1];
  const long long* batch = (const long long*)d_in[2];
  const float* W1 = (const float*)d_in[3]; const float* b1 = (const float*)d_in[4];
  const float* W2 = (const float*)d_in[5]; const float* b2 = (const float*)d_in[6];
  const float* W3 = (const float*)d_in[7]; const float* b3 = (const float*)d_in[8];
  const float* Wl = (const float*)d_in[9]; const float* bl = (const float*)d_in[10];

  const int N = in_sizes[0] / 64;     // 50000 nodes
  const int E = in_sizes[1] / 2;      // 800000 edges
  const int G = out_size;             // 512 graphs (N_CLS == 1)
  const long long* src  = ei;
  const long long* dstp = ei + E;

  // workspace layout (floats)
  float* dinv = (float*)d_ws;
  float* hbuf = dinv + (((size_t)N + 63) & ~(size_t)63);
  float* fbuf = hbuf + (size_t)N * 64;
  float* psum = fbuf + (size_t)N * 64;
  float* pcnt = psum + (size_t)G * 64;

  const int total    = N * 64;
  const int gNodes   = (N + 255) / 256;
  const int gEdges   = (E + 255) / 256;
  const int gElems   = (total + 255) / 256;
  const int gScatter = (int)(((size_t)E * 32 + 255) / 256);
  const int gPool    = (int)(((size_t)N * 32 + 255) / 256);
  const int gGemm    = ((N + 15) / 16 + 7) / 8;

  // degrees (with self-loop) -> dinv = deg^-1/2
  k_fill<<<gNodes, 256, 0, stream>>>(dinv, 1.0f, N);
  k_deg<<<gEdges, 256, 0, stream>>>(dstp, dinv, E);
  k_rsqrt<<<gNodes, 256, 0, stream>>>(dinv, N);

  // layer 1
  gcn_gemm64<<<gGemm, 256, 0, stream>>>(x, W1, hbuf, N);
  k_self<<<gElems, 256, 0, stream>>>(hbuf, dinv, fbuf, total);
  k_scatter<<<gScatter, 256, 0, stream>>>(src, dstp, dinv, hbuf, fbuf, E);
  k_finish<<<gElems, 256, 0, stream>>>(fbuf, b1, total, 1);

  // layer 2
  gcn_gemm64<<<gGemm, 256, 0, stream>>>(fbuf, W2, hbuf, N);
  k_self<<<gElems, 256, 0, stream>>>(hbuf, dinv, fbuf, total);
  k_scatter<<<gScatter, 256, 0, stream>>>(src, dstp, dinv, hbuf, fbuf, E);
  k_finish<<<gElems, 256, 0, stream>>>(fbuf, b2, total, 1);

  // layer 3 (no relu)
  gcn_gemm64<<<gGemm, 256, 0, stream>>>(fbuf, W3, hbuf, N);
  k_self<<<gElems, 256, 0, stream>>>(hbuf, dinv, fbuf, total);
  k_scatter<<<gScatter, 256, 0, stream>>>(src, dstp, dinv, hbuf, fbuf, E);
  k_finish<<<gElems, 256, 0, stream>>>(fbuf, b3, total, 0);

  // mean pool + linear head + sigmoid
  k_fill<<<(G * 64 + 255) / 256, 256, 0, stream>>>(psum, 0.0f, G * 64);
  k_fill<<<(G + 255) / 256, 256, 0, stream>>>(pcnt, 0.0f, G);
  k_pool<<<gPool, 256, 0, stream>>>(fbuf, batch, psum, pcnt, N);
  k_head<<<(G * 32 + 255) / 256, 256, 0, stream>>>(psum, pcnt, Wl, bl, (float*)d_out, G);
}